// mLSTMCell_89739046683401
// MI455X (gfx1250) — compile-verified
//
#include <hip/hip_runtime.h>
#include <hip/hip_bf16.h>

#define Bsz 64
#define Dd  2048
#define Hh  1024

typedef __attribute__((ext_vector_type(2))) float v2f;
typedef __attribute__((ext_vector_type(4))) float f4;
typedef __attribute__((ext_vector_type(8))) float v8f;

// ---------------------------------------------------------------------------
// Kernel 1: six fused projections z[g] = x @ W[g] + b[g], g in {i,f,o,q,k,v}
// FP32 WMMA 16x16x4. Each wave owns one 16(M)x16(N) tile; block = 4 waves
// covering a 16x64 tile; grid.z selects the gate.
// ---------------------------------------------------------------------------
__global__ void __launch_bounds__(128)
proj_gemm_kernel(const float* __restrict__ x,
                 const float* __restrict__ W0, const float* __restrict__ W1,
                 const float* __restrict__ W2, const float* __restrict__ W3,
                 const float* __restrict__ W4, const float* __restrict__ W5,
                 const float* __restrict__ c0, const float* __restrict__ c1,
                 const float* __restrict__ c2, const float* __restrict__ c3,
                 const float* __restrict__ c4, const float* __restrict__ c5,
                 float* __restrict__ z)
{
    const int lane = threadIdx.x & 31;
    const int wave = threadIdx.x >> 5;
    const int l    = lane & 15;        // N-index (B/C/D) or M-index (A)
    const int half = lane >> 4;        // 0: K 0..1 / M 0..7 ; 1: K 2..3 / M 8..15
    const int n0   = blockIdx.x * 64 + wave * 16;
    const int m0   = blockIdx.y * 16;
    const int g    = blockIdx.z;

    const float* W;
    const float* bias;
    switch (g) {
        case 0:  W = W0; bias = c0; break;
        case 1:  W = W1; bias = c1; break;
        case 2:  W = W2; bias = c2; break;
        case 3:  W = W3; bias = c3; break;
        case 4:  W = W4; bias = c4; break;
        default: W = W5; bias = c5; break;
    }

    const float* xrow = x + (size_t)(m0 + l) * Dd;   // A: row m0+l of x
    v8f acc = {};

    #pragma unroll 4
    for (int k = 0; k < Dd; k += 4) {
        // A (16x4): VGPR0 = K {0|2}, VGPR1 = K {1|3} per lane-half
        v2f a;
        a.x = xrow[k + 2 * half];
        a.y = xrow[k + 2 * half + 1];
        // B (4x16): row K striped across lanes; lane-half selects K pair
        const float* bp = W + (size_t)(k + 2 * half) * Hh + n0 + l;
        v2f bm;
        bm.x = bp[0];
        bm.y = bp[Hh];
        acc = __builtin_amdgcn_wmma_f32_16x16x4_f32(
                  /*neg_a=*/false, a, /*neg_b=*/false, bm,
                  /*c_mod=*/(short)0, acc, /*reuse_a=*/false, /*reuse_b=*/false);
    }

    const float bs = bias[n0 + l];
    float* zt = z + ((size_t)g * Bsz + (size_t)m0) * Hh + n0 + l;
    #pragma unroll
    for (int r = 0; r < 8; ++r) {
        const int m = r + 8 * half;            // C/D: VGPR r -> M=r (lo) / r+8 (hi)
        zt[(size_t)m * Hh] = acc[r] + bs;
    }
}

// ---------------------------------------------------------------------------
// Kernel 2: gates + stabilizer + nt + denom. One block per batch row b.
// ---------------------------------------------------------------------------
__global__ void __launch_bounds__(256)
gate_kernel(const float* __restrict__ z,
            const float* __restrict__ n_in, const float* __restrict__ m_in,
            float* __restrict__ ft_o, float* __restrict__ ot_o,
            float* __restrict__ kt_o, float* __restrict__ denom,
            float* __restrict__ nt_out, float* __restrict__ mt_out)
{
    const int b = blockIdx.x;
    const float* zi = z + ((size_t)(0 * Bsz + b)) * Hh;
    const float* zf = z + ((size_t)(1 * Bsz + b)) * Hh;
    const float* zo = z + ((size_t)(2 * Bsz + b)) * Hh;
    const float* zq = z + ((size_t)(3 * Bsz + b)) * Hh;
    const float* zk = z + ((size_t)(4 * Bsz + b)) * Hh;

    float partial = 0.0f;
    for (int h = threadIdx.x; h < Hh; h += 256) {
        const size_t idx = (size_t)b * Hh + h;
        const float ziv = zi[h];
        const float ftv = 1.0f / (1.0f + expf(-zf[h]));
        const float otv = 1.0f / (1.0f + expf(-zo[h]));
        const float ktv = zk[h] * 0.03125f;           // / sqrt(1024)
        const float mtv = fmaxf(logf(ftv) + m_in[idx], ziv);
        const float ip  = expf(ziv - mtv);
        const float ntv = fmaf(ftv, n_in[idx], ip * ktv);
        partial += ntv * zq[h];
        ft_o[idx]   = ftv;
        ot_o[idx]   = otv;
        kt_o[idx]   = ktv;
        nt_out[idx] = ntv;
        mt_out[idx] = mtv;
    }

    // wave32 + LDS reduction of 256 partials
    #pragma unroll
    for (int off = 16; off > 0; off >>= 1)
        partial += __shfl_down(partial, off, 32);
    __shared__ float sred[8];
    const int lane = threadIdx.x & 31;
    const int wid  = threadIdx.x >> 5;
    if (lane == 0) sred[wid] = partial;
    __syncthreads();
    if (threadIdx.x == 0) {
        float s = sred[0] + sred[1] + sred[2] + sred[3] +
                  sred[4] + sred[5] + sred[6] + sred[7];
        denom[b] = fmaxf(fabsf(s), 1.0f);
    }
}

// ---------------------------------------------------------------------------
// Kernel 3: ct = ft*c + v k^T, fused with h_tilde = (ct @ q)/denom, ht = ot*h.
// One 256-thread block per (b, i) row: 256 * float4 = 1024 elements.
// c/ct are streamed non-temporally (use-once, > L2).
// ---------------------------------------------------------------------------
__global__ void __launch_bounds__(256)
state_kernel(const float* __restrict__ c, const float* __restrict__ z,
             const float* __restrict__ ft, const float* __restrict__ ot,
             const float* __restrict__ kts, const float* __restrict__ denom,
             float* __restrict__ ct_out, float* __restrict__ ht_out)
{
    const int i = blockIdx.x;
    const int b = blockIdx.y;
    const int t = threadIdx.x;

    const size_t row = ((size_t)b * Hh + i) * Hh;
    const f4* crow  = (const f4*)(c + row);
    f4*       ctrow = (f4*)(ct_out + row);
    const f4* kt4   = (const f4*)(kts + (size_t)b * Hh);
    const f4* q4    = (const f4*)(z + ((size_t)(3 * Bsz + b)) * Hh);

    const float fv = ft[(size_t)b * Hh + i];
    const float vv = z[((size_t)(5 * Bsz + b)) * Hh + i];   // zv

    const f4 cv = __builtin_nontemporal_load(crow + t);
    const f4 kv = kt4[t];
    const f4 qv = q4[t];

    f4 ctv;
    ctv.x = fmaf(fv, cv.x, vv * kv.x);
    ctv.y = fmaf(fv, cv.y, vv * kv.y);
    ctv.z = fmaf(fv, cv.z, vv * kv.z);
    ctv.w = fmaf(fv, cv.w, vv * kv.w);
    __builtin_nontemporal_store(ctv, ctrow + t);

    float partial = ctv.x * qv.x + ctv.y * qv.y + ctv.z * qv.z + ctv.w * qv.w;
    #pragma unroll
    for (int off = 16; off > 0; off >>= 1)
        partial += __shfl_down(partial, off, 32);
    __shared__ float sred[8];
    const int lane = t & 31;
    const int wid  = t >> 5;
    if (lane == 0) sred[wid] = partial;
    __syncthreads();
    if (t == 0) {
        const float s = sred[0] + sred[1] + sred[2] + sred[3] +
                        sred[4] + sred[5] + sred[6] + sred[7];
        ht_out[(size_t)b * Hh + i] = ot[(size_t)b * Hh + i] * s / denom[b];
    }
}

// ---------------------------------------------------------------------------
extern "C" void kernel_launch(void* const* d_in, const int* in_sizes, int n_in,
                              void* d_out, int out_size, void* d_ws, size_t ws_size,
                              hipStream_t stream)
{
    const float* x  = (const float*)d_in[0];
    const float* c  = (const float*)d_in[1];
    const float* n  = (const float*)d_in[2];
    const float* m  = (const float*)d_in[3];
    const float* Wi = (const float*)d_in[4];
    const float* Wf = (const float*)d_in[5];
    const float* Wo = (const float*)d_in[6];
    const float* Wq = (const float*)d_in[7];
    const float* Wk = (const float*)d_in[8];
    const float* Wv = (const float*)d_in[9];
    const float* bi = (const float*)d_in[10];
    const float* bf = (const float*)d_in[11];
    const float* bo = (const float*)d_in[12];
    const float* bq = (const float*)d_in[13];
    const float* bk = (const float*)d_in[14];
    const float* bv = (const float*)d_in[15];

    float* out = (float*)d_out;               // (ht, ct, nt, mt) flat
    float* ht  = out;
    float* ct  = out + (size_t)Bsz * Hh;
    float* nt  = ct  + (size_t)Bsz * Hh * Hh;
    float* mt  = nt  + (size_t)Bsz * Hh;

    float* ws    = (float*)d_ws;
    float* z     = ws;                        // 6*B*H pre-activations
    float* ftw   = z   + (size_t)6 * Bsz * Hh;
    float* otw   = ftw + (size_t)Bsz * Hh;
    float* ktw   = otw + (size_t)Bsz * Hh;
    float* denom = ktw + (size_t)Bsz * Hh;    // B floats

    dim3 gA(Hh / 64, Bsz / 16, 6);
    proj_gemm_kernel<<<gA, 128, 0, stream>>>(x, Wi, Wf, Wo, Wq, Wk, Wv,
                                             bi, bf, bo, bq, bk, bv, z);

    gate_kernel<<<Bsz, 256, 0, stream>>>(z, n, m, ftw, otw, ktw, denom, nt, mt);

    dim3 gC(Hh, Bsz);
    state_kernel<<<gC, 256, 0, stream>>>(c, z, ftw, otw, ktw, denom, ct, ht);
}